// OffloadMultiHeadEmbedding_38517266710585
// MI455X (gfx1250) — compile-verified
//
#include <hip/hip_runtime.h>

// 128 fp32 per row -> 32 float4 chunks; one wave32 moves exactly one 512B row
// (lane l handles float4 chunk l) in a single coalesced global_load_b128.
typedef float v4f __attribute__((ext_vector_type(4)));

#define ROW_F4   32   // 128 floats / 4
#define NHEADS   8
#define UNROLL   8    // rows in flight per wave: 8 outstanding b128 loads/lane

__global__ __launch_bounds__(256) void emb_gather_kernel(
    const int* __restrict__ hash_ids,   // [B*S*H] flattened, head = idx % 8
    const v4f* __restrict__ weight,     // [TOTAL, 32] as float4
    const int* __restrict__ offsets,    // [8]
    v4f*       __restrict__ out,        // [rows, 32] as float4
    int rows)
{
    const int lane       = threadIdx.x & 31;
    const int wave       = blockIdx.x * (blockDim.x >> 5) + (threadIdx.x >> 5);
    const int totalWaves = gridDim.x * (blockDim.x >> 5);
    const int stride     = totalWaves * UNROLL;

    // Per-head base offsets (uniform, hoisted out of the loop).
    int off[NHEADS];
#pragma unroll
    for (int h = 0; h < NHEADS; ++h) off[h] = offsets[h];

    for (int base = wave * UNROLL; base < rows; base += stride) {
        // Prefetch next iteration's index cacheline (global_prefetch_b8).
        if (base + stride < rows)
            __builtin_prefetch(hash_ids + base + stride, 0, 1);

        if (base + UNROLL <= rows) {
            // ---- fast path: 8 rows per wave, loads batched before stores ----
            long gidx[UNROLL];
#pragma unroll
            for (int u = 0; u < UNROLL; ++u) {
                const int r = base + u;
                gidx[u] = (long)hash_ids[r] + (long)off[r & (NHEADS - 1)];
            }
            v4f v[UNROLL];
#pragma unroll
            for (int u = 0; u < UNROLL; ++u) {
                // Table (409MB) >> L2 (192MB), ~no reuse: non-temporal load.
                v[u] = __builtin_nontemporal_load(weight + gidx[u] * ROW_F4 + lane);
            }
#pragma unroll
            for (int u = 0; u < UNROLL; ++u) {
                // Output is write-once streamed: non-temporal store.
                __builtin_nontemporal_store(v[u], out + (long)(base + u) * ROW_F4 + lane);
            }
        } else {
            // ---- tail: per-row guarded ----
            for (int u = 0; u < UNROLL; ++u) {
                const int r = base + u;
                if (r >= rows) break;
                const long g = (long)hash_ids[r] + (long)off[r & (NHEADS - 1)];
                v4f v = __builtin_nontemporal_load(weight + g * ROW_F4 + lane);
                __builtin_nontemporal_store(v, out + (long)r * ROW_F4 + lane);
            }
        }
    }
}

extern "C" void kernel_launch(void* const* d_in, const int* in_sizes, int n_in,
                              void* d_out, int out_size, void* d_ws, size_t ws_size,
                              hipStream_t stream) {
    // setup_inputs() order: hash_ids [B,S,H] int, weight [TOTAL,128] f32, offsets [8] int
    const int* hash_ids = (const int*)d_in[0];
    const v4f* weight   = (const v4f*)d_in[1];
    const int* offsets  = (const int*)d_in[2];
    v4f*       out      = (v4f*)d_out;

    const int rows = in_sizes[0];                 // B*S*H = 131072 lookups

    const int threads       = 256;                // 8 wave32 per block
    const int wavesPerBlock = threads / 32;
    const int rowsPerBlock  = wavesPerBlock * UNROLL;   // 64 rows/block
    int blocks = (rows + rowsPerBlock - 1) / rowsPerBlock;  // 2048 blocks
    if (blocks < 1) blocks = 1;

    emb_gather_kernel<<<blocks, threads, 0, stream>>>(hash_ids, weight, offsets, out, rows);
}